// SSIMLoss_38397007626539
// MI455X (gfx1250) — compile-verified
//
#include <hip/hip_runtime.h>

// ---------------------------------------------------------------------------
// Fused SSIM loss for [32,3,512,512] fp32 via V_WMMA_F32_16X16X4_F32.
//
// One wave (32 threads) per 32x32 output macro-tile of one (b,c) plane:
//   pass 1 (vertical):   T(32x42) = Gband x In   (2 row-tiles x 3 chunks x 7)
//   pass 2 (horizontal): Out(32x32) = T x Gband  (4 out-tiles x 7)
// for the 5 maps {X, Y, X*X, Y*Y, X*Y} with shared LDS reads, then the SSIM
// map (single fast divide per pixel), masked partial sum, wave32 reduction,
// and a per-plane float atomicAdd.
//
// LDS input tiles are stored TRANSPOSED (col-major, row stride 44) so every
// fragment is one aligned ds_load_b64 into an even VGPR pair; stride 44 keeps
// the 16 fragment columns on 16 disjoint 4-bank groups (conflict-free).
// Pad rows 42/43 and pad cols 42..47 have zero band coefficients everywhere
// they are consumed, so they only need to be FINITE (zero / real data).
// ---------------------------------------------------------------------------

typedef float v2f __attribute__((ext_vector_type(2)));
typedef float v8f __attribute__((ext_vector_type(8)));

#define HW       512
#define OUTHW    502            // 512 - 11 + 1, VALID output extent
#define TPR      16             // ceil(502/32) row macro-tiles
#define TPC      16             // ceil(502/32) col macro-tiles
#define NPLANES  96             // 32 batches * 3 channels
#define SSIM_C1  (0.01f * 0.01f)
#define SSIM_C2  (0.03f * 0.03f)
#define NROW     44             // staged rows: 0..41 data, 42..43 zero K-pad
#define NCOL     48             // staged cols: 0..41 needed, 42..47 finite pad
#define TK       44             // T K-extent (cols 0..43 stored)
#define TSTRIDE  (32 * TK)      // 1408 floats per T map buffer

// Normalized Gaussian, win=11, sigma=1.5 (matches pytorch-msssim).
__constant__ float G11[11] = {
    0.00102838f, 0.00759876f, 0.03600077f, 0.10936069f, 0.21300552f,
    0.26601172f, 0.21300552f, 0.10936069f, 0.03600077f, 0.00759876f,
    0.00102838f
};

__device__ __forceinline__ float gband(int k, int i) {
    unsigned d = (unsigned)(k - i);
    return (d <= 10u) ? G11[d] : 0.0f;
}

__global__ __launch_bounds__(32)
void ssim_tile_kernel(const float* __restrict__ X,
                      const float* __restrict__ Y,
                      float* __restrict__ planeSums) {
    // Transposed halo tiles: s[col*NROW + row].
    __shared__ __align__(16) float sX[NCOL * NROW];
    __shared__ __align__(16) float sY[NCOL * NROW];
    // Pass-1 -> pass-2 staging, row-major: sT[map][row*TK + k].
    __shared__ __align__(16) float sT[5 * TSTRIDE];

    const int lane = threadIdx.x;     // 0..31 (wave32)
    const int half = lane >> 4;       // lane half selects K offset +2
    const int mn   = lane & 15;       // M for A-frags, N for B-frags / D

    const int bid   = blockIdx.x;
    const int plane = bid / (TPR * TPC);
    const int rem   = bid - plane * (TPR * TPC);
    const int tr    = rem / TPC;
    const int tc    = rem - tr * TPC;
    const int r0    = tr * 32;
    const int c0    = tc * 32;

    const float* Xp = X + (size_t)plane * HW * HW;
    const float* Yp = Y + (size_t)plane * HW * HW;

    // ---- Stage halo region (42 rows x 48 cols) into transposed LDS ------
    if (tr < TPR - 1 && tc < TPC - 1) {
        // Interior fast path: every load in-bounds, no masks.
        const size_t g0 = (size_t)(r0 + half) * HW + c0;
        {   // cols 0..31, v2f per lane, rows 2*it + half
            const int colb = mn * 2;
            const int l0 = colb * NROW + half;
            #pragma unroll
            for (int it = 0; it < 21; ++it) {
                const v2f xv = *(const v2f*)(Xp + g0 + colb + (size_t)(2 * it) * HW);
                const v2f yv = *(const v2f*)(Yp + g0 + colb + (size_t)(2 * it) * HW);
                sX[l0 + 2 * it]        = xv.x;
                sX[l0 + NROW + 2 * it] = xv.y;
                sY[l0 + 2 * it]        = yv.x;
                sY[l0 + NROW + 2 * it] = yv.y;
            }
        }
        {   // cols 32..47, scalar per lane, rows 2*it + half
            const int c2 = 32 + mn;
            const int l2 = c2 * NROW + half;
            #pragma unroll
            for (int it = 0; it < 21; ++it) {
                sX[l2 + 2 * it] = Xp[g0 + c2 + (size_t)(2 * it) * HW];
                sY[l2 + 2 * it] = Yp[g0 + c2 + (size_t)(2 * it) * HW];
            }
        }
        // Zero K-pad rows 42/43 (must be finite; coefficients are zero).
        sX[lane * NROW + 42] = 0.0f;  sX[lane * NROW + 43] = 0.0f;
        sY[lane * NROW + 42] = 0.0f;  sY[lane * NROW + 43] = 0.0f;
        sX[(32 + mn) * NROW + 42] = 0.0f;  sX[(32 + mn) * NROW + 43] = 0.0f;
        sY[(32 + mn) * NROW + 42] = 0.0f;  sY[(32 + mn) * NROW + 43] = 0.0f;
    } else {
        // Edge path (31/256 tiles): fully masked, zero padded.
        #pragma unroll
        for (int it = 0; it < NROW; ++it) {            // it == row (uniform)
            const int gr = r0 + it;
            float xv = 0.0f, yv = 0.0f;
            if (it < 42) {
                const int gc = c0 + lane;
                if (gr < HW && gc < HW) {
                    const int off = gr * HW + gc;
                    xv = Xp[off];
                    yv = Yp[off];
                }
            }
            sX[lane * NROW + it] = xv;
            sY[lane * NROW + it] = yv;
            if (lane < 16) {                           // cols 32..47
                float xv2 = 0.0f, yv2 = 0.0f;
                if (it < 42) {
                    const int gc2 = c0 + 32 + lane;
                    if (gr < HW && gc2 < HW) {
                        const int off2 = gr * HW + gc2;
                        xv2 = Xp[off2];
                        yv2 = Yp[off2];
                    }
                }
                sX[(32 + lane) * NROW + it] = xv2;
                sY[(32 + lane) * NROW + it] = yv2;
            }
        }
    }

    // ---- Gaussian band fragments (A in pass 1 == B in pass 2) -----------
    v2f gf[7];
    #pragma unroll
    for (int kb = 0; kb < 7; ++kb) {
        const int k0 = 4 * kb + 2 * half;
        gf[kb].x = gband(k0,     mn);
        gf[kb].y = gband(k0 + 1, mn);
    }

    const v8f vzero = {};

    // ---- Pass 1: vertical blur (both row-tiles share gf) ----------------
    #pragma unroll
    for (int rt = 0; rt < 2; ++rt) {
        const int rowb = rt * 16;                      // In-row / T-row base
        #pragma unroll
        for (int chunk = 0; chunk < 3; ++chunk) {
            v8f tacc[5];
            #pragma unroll
            for (int m = 0; m < 5; ++m) tacc[m] = vzero;
            const int colB = chunk * 16 + mn;          // B-frag column (N)
            #pragma unroll
            for (int kb = 0; kb < 7; ++kb) {
                const int k0 = 4 * kb + 2 * half;
                const v2f xv = *(const v2f*)(sX + colB * NROW + rowb + k0);
                const v2f yv = *(const v2f*)(sY + colB * NROW + rowb + k0);
                v2f bb[5];
                bb[0] = xv;
                bb[1] = yv;
                bb[2] = xv * xv;
                bb[3] = yv * yv;
                bb[4] = xv * yv;
                #pragma unroll
                for (int m = 0; m < 5; ++m)
                    tacc[m] = __builtin_amdgcn_wmma_f32_16x16x4_f32(
                        false, gf[kb], false, bb[m], (short)0, tacc[m], false, false);
            }
            // Scatter D (M = rowb + half*8 + r, N = colB); keep K-cols 0..43.
            #pragma unroll
            for (int m = 0; m < 5; ++m) {
                #pragma unroll
                for (int r = 0; r < 8; ++r) {
                    const int mm = rowb + half * 8 + r;
                    if (chunk < 2)
                        sT[m * TSTRIDE + mm * TK + colB] = tacc[m][r];
                    else if (mn < 12)
                        sT[m * TSTRIDE + mm * TK + colB] = tacc[m][r];
                }
            }
        }
    }

    // ---- Pass 2: horizontal blur + SSIM per 16x16 out-tile --------------
    float sum = 0.0f;
    #pragma unroll
    for (int rt = 0; rt < 2; ++rt) {
        #pragma unroll
        for (int ct = 0; ct < 2; ++ct) {
            v8f res[5];
            #pragma unroll
            for (int m = 0; m < 5; ++m) {
                v8f acc = vzero;
                #pragma unroll
                for (int kb = 0; kb < 7; ++kb) {
                    const int k0 = 4 * kb + 2 * half;
                    const v2f a = *(const v2f*)(sT + m * TSTRIDE
                                                + (rt * 16 + mn) * TK
                                                + ct * 16 + k0);
                    acc = __builtin_amdgcn_wmma_f32_16x16x4_f32(
                        false, a, false, gf[kb], (short)0, acc, false, false);
                }
                res[m] = acc;
            }
            const bool colOK = (c0 + ct * 16 + mn) < OUTHW;
            #pragma unroll
            for (int r = 0; r < 8; ++r) {
                const int orow = r0 + rt * 16 + half * 8 + r;
                const float mu1 = res[0][r];
                const float mu2 = res[1][r];
                const float mu1sq = mu1 * mu1;
                const float mu2sq = mu2 * mu2;
                const float mu12  = mu1 * mu2;
                const float s1  = res[2][r] - mu1sq;
                const float s2  = res[3][r] - mu2sq;
                const float s12 = res[4][r] - mu12;
                // ((2*mu12+C1)/(mu1sq+mu2sq+C1)) * ((2*s12+C2)/(s1+s2+C2))
                const float num = (2.0f * mu12 + SSIM_C1) * (2.0f * s12 + SSIM_C2);
                const float den = (mu1sq + mu2sq + SSIM_C1) * (s1 + s2 + SSIM_C2);
                const float ss  = __fdividef(num, den);
                const bool valid = colOK && orow < OUTHW;
                sum += valid ? ss : 0.0f;
            }
        }
    }

    // ---- wave32 reduction + per-plane accumulation ----------------------
    #pragma unroll
    for (int off = 16; off > 0; off >>= 1)
        sum += __shfl_down(sum, off, 32);
    if (lane == 0)
        atomicAdd(&planeSums[plane], sum);
}

__global__ void ssim_zero_ws(float* ws) {
    if (threadIdx.x < NPLANES) ws[threadIdx.x] = 0.0f;
}

__global__ void ssim_finalize(const float* __restrict__ ws, float* __restrict__ out) {
    const float inv = 1.0f / ((float)OUTHW * (float)OUTHW);
    float s = 0.0f;
    for (int i = 0; i < NPLANES; ++i)
        s += fmaxf(ws[i] * inv, 0.0f);           // per-channel mean + relu
    out[0] = 1.0f - s / (float)NPLANES;          // 1 - mean over (B, C)
}

extern "C" void kernel_launch(void* const* d_in, const int* in_sizes, int n_in,
                              void* d_out, int out_size, void* d_ws, size_t ws_size,
                              hipStream_t stream) {
    (void)in_sizes; (void)n_in; (void)out_size; (void)ws_size;
    const float* X  = (const float*)d_in[0];   // predictions
    const float* Y  = (const float*)d_in[1];   // labels
    float* ws  = (float*)d_ws;                 // 96 per-plane partial sums
    float* out = (float*)d_out;                // scalar fp32 loss

    ssim_zero_ws<<<1, 128, 0, stream>>>(ws);
    ssim_tile_kernel<<<dim3(NPLANES * TPR * TPC), 32, 0, stream>>>(X, Y, ws);
    ssim_finalize<<<1, 1, 0, stream>>>(ws, out);
}